// Graph_Embedding_59004260712652
// MI455X (gfx1250) — compile-verified
//
#include <hip/hip_runtime.h>
#include <math.h>

typedef __attribute__((ext_vector_type(2))) float v2f;
typedef __attribute__((ext_vector_type(4))) float v4f;
typedef __attribute__((ext_vector_type(8))) float v8f;

#define NV 10   // node count V
#define NF 256  // feature dim F

#define LOG2E 1.4426950408889634f

// ---------------------------------------------------------------------------
// Kernel A: one wave32 per (b,t).
//   edge_feature = tanh(sigmoid(gauss(Ssym @ mul + bias)))   [WMMA f32 16x16x4]
//   Di (degree indices) -> workspace for kernel B
//   Floyd-Warshall in registers (row per lane, readlane broadcast)
//   atten_bias = sum_j emb3[dist_i] + emb4[sp_i]
// ---------------------------------------------------------------------------
__global__ __launch_bounds__(32)
void graph_small_kernel(const float* __restrict__ S_g,
                        const float* __restrict__ mul_g,
                        const float* __restrict__ bias_g,
                        const float* __restrict__ means_g,
                        const float* __restrict__ stds_g,
                        const float* __restrict__ emb3_g,
                        const float* __restrict__ emb4_g,
                        float* __restrict__ atten_bias,  // (BT,10,10)
                        int* __restrict__ Di_out)        // (BT,10)
{
    __shared__ float sA[16 * 12];      // symmetrized S, zero padded (WMMA A)
    __shared__ float sBt[16 * 12];     // mul^T, zero padded        (WMMA B)
    __shared__ float sEFp[16 * 16];    // edge feature, padded tile
    __shared__ float sBiasT[16 * 16];  // bias^T, padded
    __shared__ float sMean[16];
    __shared__ float sStd[16];
    __shared__ float sE3[112];
    __shared__ float sE4[112];
    __shared__ int   sSIi[16 * 12];    // also float temp for raw mul
    __shared__ int   sEIi[16 * 12];    // also float temp for raw bias
    __shared__ float sRaw[112];

    const int  bt   = blockIdx.x;
    const int  lane = threadIdx.x;
    const long base = (long)bt * (NV * NV);

    // ---- phase 1: zero pads + bulk float4 loads -----------------------------
    {
        v4f  z  = {0.0f, 0.0f, 0.0f, 0.0f};
        v4f* zA = (v4f*)sA;
        v4f* zB = (v4f*)sBt;
        zA[lane] = z;
        zB[lane] = z;
        if (lane < 16) { zA[lane + 32] = z; zB[lane + 32] = z; }
        if (lane < 25) {
            ((v4f*)sRaw)[lane] = ((const v4f*)(S_g + base))[lane];
            ((v4f*)sSIi)[lane] = ((const v4f*)mul_g)[lane];   // raw mul (temp)
            ((v4f*)sEIi)[lane] = ((const v4f*)bias_g)[lane];  // raw bias (temp)
            ((v4f*)sE3)[lane]  = ((const v4f*)emb3_g)[lane];
            ((v4f*)sE4)[lane]  = ((const v4f*)emb4_g)[lane];
        }
        if (lane < 16) {
            sMean[lane] = (lane < NV) ? means_g[lane] : 0.0f;
            sStd[lane]  = (lane < NV) ? stds_g[lane]  : 1.0f;
        }
    }
    __syncthreads();

    // ---- phase 2: symmetrize S; transpose mul and bias into padded tiles ----
    {
        const float* tmpM = (const float*)sSIi;
        const float* tmpB = (const float*)sEIi;
        for (int e = lane; e < NV * NV; e += 32) {
            int i = e / NV, j = e - i * NV;
            sA[i * 12 + j]     = fminf(sRaw[e], sRaw[j * NV + i]);
            sBt[j * 12 + i]    = tmpM[e];  // Bt[n][k] = mul[k][n]
            sBiasT[j * 16 + i] = tmpB[e];  // BiasT[n][m] = bias[m][n]
        }
    }
    __syncthreads();

    // ---- x = Ssym @ mul via V_WMMA_F32_16X16X4_F32 (3 K-steps, K 10->12) ----
    // A 16x4 f32: lanes 0-15 M=lane (v0:K+0, v1:K+1); lanes 16-31 (v0:K+2, v1:K+3)
    // B 4x16: symmetric analog with N=lane&15 (loaded from mul^T so pair is contiguous)
    const int mn  = lane & 15;
    const int hi  = lane >> 4;
    const int khi = hi * 2;
    v8f c = {0.f, 0.f, 0.f, 0.f, 0.f, 0.f, 0.f, 0.f};
    {
        const v2f* pA = (const v2f*)&sA[mn * 12 + khi];
        const v2f* pB = (const v2f*)&sBt[mn * 12 + khi];
#pragma unroll
        for (int kk = 0; kk < 3; ++kk) {
            v2f a = pA[kk * 2];
            v2f b = pB[kk * 2];
            c = __builtin_amdgcn_wmma_f32_16x16x4_f32(
                    false, a, false, b, (short)0, c, false, false);
        }
    }

    // ---- Gaussian -> sigmoid -> tanh, branch-free, hardware rcp/exp2 --------
    {
        const float mean = sMean[mn];
        const float inv  = __builtin_amdgcn_rcpf(sStd[mn]);   // 1/sd (v_rcp_f32)
        const float gg   = inv * 0.39894229f;                 // 1/(sqrt(2*3.14159)*sd)
        const int   m0   = hi * 8;                            // M = r + 8*(lane/16)
        const float* pBias = &sBiasT[mn * 16 + m0];
        float*       pEF   = &sEFp[m0 * 16 + mn];
#pragma unroll
        for (int r = 0; r < 8; ++r) {
            float x   = c[r] + pBias[r];
            float t   = (x - mean) * inv;
            // exp(-0.5*t^2)/(a*sd) via exp2 with folded log2e
            float tmp = __builtin_amdgcn_exp2f(t * t * (-0.5f * LOG2E)) * gg;
            // sigmoid(tmp) = 1/(1+exp(-tmp))
            float sig = __builtin_amdgcn_rcpf(
                            1.0f + __builtin_amdgcn_exp2f(-LOG2E * tmp));
            // tanh(sig) = 1 - 2/(exp(2*sig)+1), sig in (0.5,1) so well conditioned
            float e2  = __builtin_amdgcn_exp2f((2.0f * LOG2E) * sig);
            pEF[r * 16] = 1.0f - 2.0f * __builtin_amdgcn_rcpf(e2 + 1.0f);
        }
    }

    // ---- Degree indices: D[j] = sum_i Ssym[i][j]; Di = int(D) ---------------
    {
        const int col = (lane < NV) ? lane : 0;
        float D = 0.0f;
#pragma unroll
        for (int i = 0; i < NV; ++i) D += sA[i * 12 + col];
        if (lane < NV) {
            int di = (int)D;
            di = di < 0 ? 0 : (di > NF - 1 ? NF - 1 : di);
            Di_out[(long)bt * NV + lane] = di;
        }
    }
    __syncthreads();  // sEFp complete before FW reads it

    // ---- Floyd-Warshall: lane i owns row i (lanes>=10 shadow row 0) ---------
    const int row = (lane < NV) ? lane : 0;
    float d[NV], sp[NV], ef[NV];
#pragma unroll
    for (int j = 0; j < NV; ++j) {
        d[j]  = sA[row * 12 + j];
        ef[j] = sEFp[row * 16 + j];
        sp[j] = 0.0f;
    }
#pragma unroll
    for (int k = 0; k < NV; ++k) {
        float dik = d[k];
        float rk[NV];
#pragma unroll
        for (int j = 0; j < NV; ++j)
            rk[j] = __int_as_float(__builtin_amdgcn_readlane(__float_as_int(d[j]), k));
#pragma unroll
        for (int j = 0; j < NV; ++j) {
            float temp = dik + rk[j];
            bool  u    = temp < d[j];         // upd = (min(dist,temp) != dist)
            sp[j] = u ? sp[j] + ef[j] : sp[j];
            d[j]  = u ? temp : d[j];
        }
    }
    if (lane < NV) {
#pragma unroll
        for (int j = 0; j < NV; ++j) {
            int a = (int)d[j];  a = a < 0 ? 0 : (a > NV - 1 ? NV - 1 : a);
            int b = (int)sp[j]; b = b < 0 ? 0 : (b > NV - 1 ? NV - 1 : b);
            sSIi[lane * 12 + j] = a;
            sEIi[lane * 12 + j] = b;
        }
    }
    __syncthreads();

    // ---- atten_bias[i][c] = sum_j emb3[SI[i][j]][c] + emb4[EI[i][j]][c] -----
#pragma unroll
    for (int ee = 0; ee < 4; ++ee) {
        int e = lane + ee * 32;
        if (e < NV * NV) {
            int i  = e / NV;
            int cc = e - i * NV;
            const int* sip = &sSIi[i * 12];
            const int* eip = &sEIi[i * 12];
            float acc = 0.0f;
#pragma unroll
            for (int j = 0; j < NV; ++j)
                acc += sE3[sip[j] * NV + cc] + sE4[eip[j] * NV + cc];
            atten_bias[base + e] = acc;
        }
    }
}

// ---------------------------------------------------------------------------
// Kernel B: streaming  out = end_output + emb_in[Di] + emb_out[Di]
// float4-vectorized; 256KB emb tables stay in L2; HBM-bound (~524MB traffic).
// ---------------------------------------------------------------------------
__global__ __launch_bounds__(256)
void add_degree_emb_kernel(const float4* __restrict__ end_output,
                           const float4* __restrict__ emb_in,
                           const float4* __restrict__ emb_out,
                           const int* __restrict__ Di,
                           float4* __restrict__ out,
                           long n4)
{
    long g = (long)blockIdx.x * blockDim.x + threadIdx.x;
    if (g >= n4) return;
    int  rv  = (int)(g >> 6);  // (b,t,v) row: 64 float4 per F=256 row
    int  f4i = (int)(g & 63);
    int  di  = Di[rv];
    float4 e = end_output[g];
    float4 a = emb_in[di * 64 + f4i];
    float4 b = emb_out[di * 64 + f4i];
    e.x += a.x + b.x;
    e.y += a.y + b.y;
    e.z += a.z + b.z;
    e.w += a.w + b.w;
    out[g] = e;
}

// ---------------------------------------------------------------------------
extern "C" void kernel_launch(void* const* d_in, const int* in_sizes, int n_in,
                              void* d_out, int out_size, void* d_ws, size_t ws_size,
                              hipStream_t stream)
{
    const float* end_output = (const float*)d_in[0];
    const float* S          = (const float*)d_in[1];
    const float* mul        = (const float*)d_in[2];
    const float* bias       = (const float*)d_in[3];
    const float* means      = (const float*)d_in[4];
    const float* stds       = (const float*)d_in[5];
    const float* emb_in     = (const float*)d_in[6];
    const float* emb_out    = (const float*)d_in[7];
    const float* emb3       = (const float*)d_in[8];
    const float* emb4       = (const float*)d_in[9];

    const int BT = in_sizes[1] / (NV * NV);  // 25600

    float* out_main = (float*)d_out;                 // (BT, V, F)
    float* atten    = out_main + (long)BT * NV * NF; // (BT, V, V)
    int*   Di       = (int*)d_ws;                    // (BT, V)

    graph_small_kernel<<<BT, 32, 0, stream>>>(
        S, mul, bias, means, stds, emb3, emb4, atten, Di);

    long n4     = (long)BT * NV * (NF / 4);
    int  blocks = (int)((n4 + 255) / 256);
    add_degree_emb_kernel<<<blocks, 256, 0, stream>>>(
        (const float4*)end_output, (const float4*)emb_in,
        (const float4*)emb_out, Di, (float4*)out_main, n4);
}